// ProteinSpecificLoss_4088808866019
// MI455X (gfx1250) — compile-verified
//
#include <hip/hip_runtime.h>

// ---------------------------------------------------------------------------
// SSIM(7x7 uniform) + MSE + symmetry loss for [128,1,384,384] fp32.
// Box sums computed on the CDNA5 matrix pipe: Box7x7(Q) = Band * Q * Band^T
// via chained V_WMMA_F32_16X16X4_F32 (K=4 chunks, accumulate through C).
// One wave (32 lanes) per 16x16 output tile; 22x22 input window in LDS.
// ---------------------------------------------------------------------------

typedef __attribute__((ext_vector_type(2))) float v2f;
typedef __attribute__((ext_vector_type(8))) float v8f;

#define HH     384
#define BB     128
#define OUTD   378          // 384 - 6 valid outputs per dim
#define NTILE  24           // ceil(378/16)
#define WEXT   22           // 16 + 6 window extent
#define NPART  (BB * NTILE * NTILE)   // 73728 partial triples

__device__ __forceinline__ float bandv(int m, int k) {
    // Banded ones matrix: 1.0 iff m <= k <= m+6  (7-tap box)
    return (k >= m && k <= m + 6) ? 1.0f : 0.0f;
}

__global__ __launch_bounds__(32)
void ssim_tile_kernel(const float* __restrict__ pred,
                      const float* __restrict__ target,
                      float* __restrict__ partials)
{
    __shared__ float su[WEXT][24];   // raw pred window (22x22, padded stride)
    __shared__ float sv[WEXT][24];   // raw target window
    __shared__ float vls[16][33];    // vertical-sum intermediate V (16x24 used)

    const int tx = blockIdx.x, ty = blockIdx.y, b = blockIdx.z;
    const int lane = threadIdx.x;          // 0..31, one wave
    const int nsub = lane & 15;
    const int half = lane >> 4;
    const int r0 = ty * 16, c0 = tx * 16;

    const float* pb = pred   + (size_t)b * HH * HH;
    const float* tb = target + (size_t)b * HH * HH;

    // ---- Load 22x22 window (clamped at image edge; clamped values are
    //      band-masked out of any valid SSIM output) ----
    for (int e = lane; e < WEXT * WEXT; e += 32) {
        const int r = e / WEXT, c = e % WEXT;
        const int gr = min(r0 + r, HH - 1);
        const int gc = min(c0 + c, HH - 1);
        su[r][c] = pb[gr * HH + gc];
        sv[r][c] = tb[gr * HH + gc];
    }
    __syncthreads();

    // ---- MSE + symmetry over the exclusively-owned 16x16 input block ----
    float mseAcc = 0.f, symAcc = 0.f, ssimAcc = 0.f;
#pragma unroll
    for (int k = 0; k < 8; ++k) {
        const int e = lane + 32 * k;
        const int r = e >> 4, c = e & 15;
        const float u = su[r][c], v = sv[r][c];
        const float d = u - v;
        mseAcc += d * d;
        const int gi = r0 + r, gj = c0 + c;       // transposed read: L2-resident
        const float dt = u - pb[gj * HH + gi];
        symAcc += dt * dt;
    }

    // ---- Box sums of {x, y, x*x, y*y, x*y} via WMMA band products ----
    v8f Sq[5];
    for (int q = 0; q < 5; ++q) {
        // Step 1: V(16x22) = Band(16x24) * Qpad(24x22)   (two 16-col chunks)
        v8f vt0 = {0.f,0.f,0.f,0.f,0.f,0.f,0.f,0.f};
        v8f vt1 = {0.f,0.f,0.f,0.f,0.f,0.f,0.f,0.f};
#pragma unroll
        for (int c4 = 0; c4 < 6; ++c4) {
            const int k0 = 4 * c4 + 2 * half;          // K rows for this lane-half
            v2f A;                                      // A: 16x4 band slice
            A.x = bandv(nsub, k0);
            A.y = bandv(nsub, k0 + 1);
            const int rr0 = min(k0, WEXT - 1);
            const int rr1 = min(k0 + 1, WEXT - 1);      // padded K rows band to 0
            // chunk 0: window cols 0..15
            {
                const int col = nsub;
                const float u0 = su[rr0][col], w0 = sv[rr0][col];
                const float u1 = su[rr1][col], w1 = sv[rr1][col];
                v2f Bv;
                Bv.x = (q==0)?u0:(q==1)?w0:(q==2)?u0*u0:(q==3)?w0*w0:u0*w0;
                Bv.y = (q==0)?u1:(q==1)?w1:(q==2)?u1*u1:(q==3)?w1*w1:u1*w1;
                vt0 = __builtin_amdgcn_wmma_f32_16x16x4_f32(
                        false, A, false, Bv, (short)0, vt0, false, false);
            }
            // chunk 1: window cols 16..21 (cols >=22 garbage, band-masked later)
            {
                const int col = min(16 + nsub, WEXT - 1);
                const float u0 = su[rr0][col], w0 = sv[rr0][col];
                const float u1 = su[rr1][col], w1 = sv[rr1][col];
                v2f Bv;
                Bv.x = (q==0)?u0:(q==1)?w0:(q==2)?u0*u0:(q==3)?w0*w0:u0*w0;
                Bv.y = (q==0)?u1:(q==1)?w1:(q==2)?u1*u1:(q==3)?w1*w1:u1*w1;
                vt1 = __builtin_amdgcn_wmma_f32_16x16x4_f32(
                        false, A, false, Bv, (short)0, vt1, false, false);
            }
        }
        __syncthreads();   // previous iteration's step-2 reads of vls done
        // D-layout -> LDS (converts to A-layout for step 2)
#pragma unroll
        for (int p = 0; p < 8; ++p) {
            vls[p + 8 * half][nsub]      = vt0[p];
            vls[p + 8 * half][16 + nsub] = vt1[p];
        }
        __syncthreads();
        // Step 2: Out(16x16) = V(16x24) * Band(24x16)
        v8f acc = {0.f,0.f,0.f,0.f,0.f,0.f,0.f,0.f};
#pragma unroll
        for (int c4 = 0; c4 < 6; ++c4) {
            const int k0 = 4 * c4 + 2 * half;
            v2f A, Bv;
            A.x = vls[nsub][k0];          // A: V slice, M=nsub, K=k0..k0+1
            A.y = vls[nsub][k0 + 1];
            Bv.x = bandv(nsub, k0);       // B: band^T slice, N=nsub
            Bv.y = bandv(nsub, k0 + 1);
            acc = __builtin_amdgcn_wmma_f32_16x16x4_f32(
                    false, A, false, Bv, (short)0, acc, false, false);
        }
        Sq[q] = acc;
    }

    // ---- Pointwise SSIM on the 5 co-laid-out D tiles ----
    const float inv49 = 1.0f / 49.0f;
    const float inv48 = 1.0f / 48.0f;
    const float K1 = 1e-4f;   // (0.01*range)^2
    const float K2 = 9e-4f;   // (0.03*range)^2
#pragma unroll
    for (int p = 0; p < 8; ++p) {
        const int orow = r0 + p + 8 * half;
        const int ocol = c0 + nsub;
        const float s1  = Sq[0][p], s2  = Sq[1][p];
        const float s11 = Sq[2][p], s22 = Sq[3][p], s12 = Sq[4][p];
        const float ux = s1 * inv49, uy = s2 * inv49;
        const float vx  = (s11 - s1 * ux) * inv48;   // cov-normalized
        const float vy  = (s22 - s2 * uy) * inv48;
        const float vxy = (s12 - s1 * uy) * inv48;
        const float A1 = 2.f * ux * uy + K1;
        const float A2 = 2.f * vxy + K2;
        const float B1 = ux * ux + uy * uy + K1;
        const float B2 = vx + vy + K2;
        const float S = (A1 * A2) * __builtin_amdgcn_rcpf(B1 * B2);
        if (orow < OUTD && ocol < OUTD) ssimAcc += S;
    }

    // ---- Wave reduce (32 lanes) -> per-block partial triple ----
    for (int off = 16; off > 0; off >>= 1) {
        mseAcc  += __shfl_xor(mseAcc,  off, 32);
        symAcc  += __shfl_xor(symAcc,  off, 32);
        ssimAcc += __shfl_xor(ssimAcc, off, 32);
    }
    if (lane == 0) {
        const int idx = ((b * NTILE + ty) * NTILE + tx) * 3;
        partials[idx + 0] = mseAcc;
        partials[idx + 1] = symAcc;
        partials[idx + 2] = ssimAcc;
    }
}

// ---- Deterministic two-stage tree reduction (no float atomics) ----
__global__ __launch_bounds__(256)
void reduce1(const float* __restrict__ partials, float* __restrict__ out2)
{
    __shared__ float sm[256][3];
    const int t = threadIdx.x, blk = blockIdx.x;
    const int CH = NPART / 256;                 // 288 triples per block
    float a0 = 0.f, a1 = 0.f, a2 = 0.f;
    const int base = blk * CH;
    for (int j = t; j < CH; j += 256) {
        const int idx = (base + j) * 3;
        a0 += partials[idx]; a1 += partials[idx + 1]; a2 += partials[idx + 2];
    }
    sm[t][0] = a0; sm[t][1] = a1; sm[t][2] = a2;
    __syncthreads();
    for (int s = 128; s > 0; s >>= 1) {
        if (t < s) {
            sm[t][0] += sm[t + s][0];
            sm[t][1] += sm[t + s][1];
            sm[t][2] += sm[t + s][2];
        }
        __syncthreads();
    }
    if (t == 0) {
        const int o = blk * 3;
        out2[o] = sm[0][0]; out2[o + 1] = sm[0][1]; out2[o + 2] = sm[0][2];
    }
}

__global__ __launch_bounds__(256)
void reduce2(const float* __restrict__ out2, float* __restrict__ out)
{
    __shared__ float sm[256][3];
    const int t = threadIdx.x;
    const int idx = t * 3;
    sm[t][0] = out2[idx]; sm[t][1] = out2[idx + 1]; sm[t][2] = out2[idx + 2];
    __syncthreads();
    for (int s = 128; s > 0; s >>= 1) {
        if (t < s) {
            sm[t][0] += sm[t + s][0];
            sm[t][1] += sm[t + s][1];
            sm[t][2] += sm[t + s][2];
        }
        __syncthreads();
    }
    if (t == 0) {
        const float npix = (float)BB * (float)HH * (float)HH;     // 18874368
        const float nout = (float)BB * (float)OUTD * (float)OUTD; // 128*378*378
        const float mse = sm[0][0] / npix;
        const float sym = sm[0][1] / npix;
        const float ssim_loss = 1.0f - sm[0][2] / nout;
        const float total = mse + 0.3f * ssim_loss + 0.1f * sym;
        out[0] = total; out[1] = mse; out[2] = ssim_loss; out[3] = sym;
    }
}

extern "C" void kernel_launch(void* const* d_in, const int* in_sizes, int n_in,
                              void* d_out, int out_size, void* d_ws, size_t ws_size,
                              hipStream_t stream)
{
    const float* pred   = (const float*)d_in[0];
    const float* target = (const float*)d_in[1];
    float* out = (float*)d_out;
    float* ws  = (float*)d_ws;

    float* partials = ws;                     // NPART * 3 floats (~884 KB)
    float* stage2   = ws + (size_t)NPART * 3; // 256 * 3 floats

    dim3 grid(NTILE, NTILE, BB);              // 24 x 24 x 128 one-wave tiles
    ssim_tile_kernel<<<grid, 32, 0, stream>>>(pred, target, partials);
    reduce1<<<256, 256, 0, stream>>>(partials, stage2);
    reduce2<<<1, 256, 0, stream>>>(stage2, out);
}